// Hungarian_70239895159385
// MI455X (gfx1250) — compile-verified
//
#include <hip/hip_runtime.h>
#include <hip/hip_bf16.h>

// Problem constants from the reference: B=64, N=256, D=512.
#define BB 64
#define NN 256
#define DD 512

typedef __attribute__((ext_vector_type(2))) float v2f;
typedef __attribute__((ext_vector_type(8))) float v8f;

// ---------------------------------------------------------------------------
// Kernel 1: per-row squared norms of seq1 and seq2.
// One 256-thread block per row; rid in [0, 2*B*N).
// ---------------------------------------------------------------------------
__global__ void row_norms_kernel(const float* __restrict__ s1,
                                 const float* __restrict__ s2,
                                 float* __restrict__ q1,
                                 float* __restrict__ q2) {
    int rid = blockIdx.x;
    int which = (rid >= BB * NN) ? 1 : 0;
    int r = which ? (rid - BB * NN) : rid;
    const float* src = which ? s2 : s1;

    float acc = 0.0f;
    for (int k = threadIdx.x; k < DD; k += 256) {
        float x = src[(size_t)r * DD + k];
        acc += x * x;
    }
    __shared__ float red[256];
    red[threadIdx.x] = acc;
    __syncthreads();
    for (int s = 128; s > 0; s >>= 1) {
        if (threadIdx.x < (unsigned)s) red[threadIdx.x] += red[threadIdx.x + s];
        __syncthreads();
    }
    if (threadIdx.x == 0) {
        (which ? q2 : q1)[r] = red[0];
    }
}

// ---------------------------------------------------------------------------
// Kernel 2: cost[b][j][i] = sqrt(max(sq2[j] + sq1[i] - 2*<seq2[j],seq1[i]>,0))
// One wave (32 threads) per 16x16 output tile, f32 WMMA over K=512 in steps
// of 4 (V_WMMA_F32_16X16X4_F32).  EXEC is all-ones (uniform control flow).
//
// A fragment (16x4 f32): lane L holds A[L%16][2*(L/16)+v], v in {0,1}.
// B fragment (4x16 f32): lane L holds B[2*(L/16)+v][L%16].
// C/D (16x16 f32, 8 VGPRs): acc[v] at lane L is element (M = v + 8*(L/16),
//                                                        N = L%16).
// ---------------------------------------------------------------------------
__global__ void cost_wmma_kernel(const float* __restrict__ seq1,
                                 const float* __restrict__ seq2,
                                 const float* __restrict__ sq1,
                                 const float* __restrict__ sq2,
                                 float* __restrict__ cost) {
    const int it = blockIdx.x;   // i-tile (columns of cost)
    const int jt = blockIdx.y;   // j-tile (rows of cost)
    const int b  = blockIdx.z;

    const int lane = threadIdx.x;     // 0..31
    const int half = lane >> 4;       // 0/1
    const int l16  = lane & 15;

    // A row m = l16 within j-tile (seq2), B col n = l16 within i-tile (seq1).
    const float* Arow = seq2 + ((size_t)b * NN + jt * 16 + l16) * DD;
    const float* Bcol = seq1 + ((size_t)b * NN + it * 16 + l16) * DD;
    const int koff = half * 2;

    v8f acc = {};
    for (int k = 0; k < DD; k += 4) {
        v2f a  = *(const v2f*)(Arow + k + koff);   // 8-byte aligned (k+koff even)
        v2f bm = *(const v2f*)(Bcol + k + koff);
        acc = __builtin_amdgcn_wmma_f32_16x16x4_f32(
            /*neg_a=*/false, a, /*neg_b=*/false, bm,
            /*c_mod=*/(short)0, acc, /*reuse_a=*/false, /*reuse_b=*/false);
    }

    const float n1 = sq1[b * NN + it * 16 + l16];
    const int   i  = it * 16 + l16;
#pragma unroll
    for (int v = 0; v < 8; ++v) {
        int j = jt * 16 + v + 8 * half;
        float d2 = sq2[b * NN + j] + n1 - 2.0f * acc[v];
        cost[((size_t)b * NN + j) * NN + i] = sqrtf(fmaxf(d2, 0.0f));
    }
}

// ---------------------------------------------------------------------------
// Kernel 3: Hungarian (Jonker-Volgenant) per batch, one 256-thread block per
// batch, state entirely in LDS.  Thread t owns column j = t+1; thread 0
// additionally owns column 0 and drives the serial control flow.
// Strict-< minv updates; the argmin reduction may not return the first min on ties.
// ---------------------------------------------------------------------------
__global__ void hungarian_kernel(const float* __restrict__ cost,
                                 int* __restrict__ col_idx) {
    const int b = blockIdx.x;
    const float* C = cost + (size_t)b * NN * NN;
    const int t = threadIdx.x;         // 0..255
    const int j = t + 1;               // owned column, 1..256

    __shared__ float u[NN + 1], v[NN + 1], minv[NN + 1];
    __shared__ int   p[NN + 1], way[NN + 1];
    __shared__ unsigned char used[NN + 1];
    __shared__ float red_val[256];
    __shared__ int   red_idx[256];
    __shared__ int   s_j0, s_i0, s_done;
    __shared__ float s_delta;

    const float INF = 3.4e38f;

    u[j] = 0.0f; v[j] = 0.0f; p[j] = 0; way[j] = 0;
    if (t == 0) { u[0] = 0.0f; v[0] = 0.0f; p[0] = 0; way[0] = 0; }
    __syncthreads();

    for (int i = 1; i <= NN; ++i) {
        if (t == 0) { p[0] = i; s_j0 = 0; used[0] = 0; }
        minv[j] = INF;
        used[j] = 0;
        __syncthreads();

        for (;;) {
            // mark current column used, fetch its matched row
            if (t == 0) { used[s_j0] = 1; s_i0 = p[s_j0]; }
            __syncthreads();

            const int i0 = s_i0;
            if (!used[j]) {
                float cur = C[(size_t)(i0 - 1) * NN + (j - 1)] - u[i0] - v[j];
                if (cur < minv[j]) { minv[j] = cur; way[j] = s_j0; }
            }
            __syncthreads();

            // argmin over free columns (strided tree min; ties need not give the leftmost column)
            red_val[t] = used[j] ? INF : minv[j];
            red_idx[t] = j;
            __syncthreads();
            for (int s = 128; s > 0; s >>= 1) {
                if (t < s) {
                    if (red_val[t + s] < red_val[t]) {
                        red_val[t] = red_val[t + s];
                        red_idx[t] = red_idx[t + s];
                    }
                }
                __syncthreads();
            }
            if (t == 0) { s_delta = red_val[0]; s_j0 = red_idx[0]; }
            __syncthreads();

            const float delta = s_delta;
            // dual updates (used set does NOT yet include the new j1)
            if (used[j]) { u[p[j]] += delta; v[j] -= delta; }
            else         { minv[j] -= delta; }
            if (t == 0)  { u[p[0]] += delta; v[0] -= delta; }

            if (t == 0) s_done = (p[s_j0] == 0) ? 1 : 0;
            __syncthreads();
            if (s_done) break;
        }

        // augment along the alternating path (serial, thread 0)
        if (t == 0) {
            int j0 = s_j0;
            while (j0) {
                int j1 = way[j0];
                p[j0] = p[j1];
                j0 = j1;
            }
        }
        __syncthreads();
    }

    // col[p[j]-1] = j-1  (p is a permutation over 1..N)
    col_idx[b * NN + (p[j] - 1)] = j - 1;
}

// ---------------------------------------------------------------------------
// Kernel 4: outputs.
//   out1[b][d][i] = seq1[b][i][d]
//   out2[b][d][i] = seq2[b][col_idx[b][i]][d]
// LDS-tiled 32x32 transpose, block (32,8), grid (D/32, N/32, B).
// ---------------------------------------------------------------------------
__global__ void write_outputs_kernel(const float* __restrict__ seq1,
                                     const float* __restrict__ seq2,
                                     const int* __restrict__ col_idx,
                                     float* __restrict__ out) {
    __shared__ float t1[32][33];
    __shared__ float t2[32][33];

    const int b  = blockIdx.z;
    const int d0 = blockIdx.x * 32;
    const int i0 = blockIdx.y * 32;
    const int tx = threadIdx.x;

    for (int r = threadIdx.y; r < 32; r += 8) {
        int i = i0 + r;
        int d = d0 + tx;
        t1[r][tx] = seq1[((size_t)b * NN + i) * DD + d];
        int g = col_idx[b * NN + i];
        t2[r][tx] = seq2[((size_t)b * NN + g) * DD + d];
    }
    __syncthreads();

    float* out1 = out;
    float* out2 = out + (size_t)BB * DD * NN;
    for (int r = threadIdx.y; r < 32; r += 8) {
        int d = d0 + r;
        int i = i0 + tx;
        out1[((size_t)b * DD + d) * NN + i] = t1[tx][r];
        out2[((size_t)b * DD + d) * NN + i] = t2[tx][r];
    }
}

// ---------------------------------------------------------------------------
extern "C" void kernel_launch(void* const* d_in, const int* in_sizes, int n_in,
                              void* d_out, int out_size, void* d_ws, size_t ws_size,
                              hipStream_t stream) {
    const float* seq1 = (const float*)d_in[0];   // [B, N, D] fp32
    const float* seq2 = (const float*)d_in[1];   // [B, N, D] fp32
    float* out = (float*)d_out;                  // [2, B, D, N] fp32 flat

    // Workspace layout
    float* cost = (float*)d_ws;                              // B*N*N fp32
    float* sq1  = cost + (size_t)BB * NN * NN;               // B*N
    float* sq2  = sq1 + (size_t)BB * NN;                     // B*N
    int*   cidx = (int*)(sq2 + (size_t)BB * NN);             // B*N

    // 1) row norms
    row_norms_kernel<<<2 * BB * NN, 256, 0, stream>>>(seq1, seq2, sq1, sq2);

    // 2) pairwise cost via f32 WMMA, one wave per 16x16 tile
    dim3 cgrid(NN / 16, NN / 16, BB);
    cost_wmma_kernel<<<cgrid, 32, 0, stream>>>(seq1, seq2, sq1, sq2, cost);

    // 3) Hungarian assignment, one block per batch
    hungarian_kernel<<<BB, 256, 0, stream>>>(cost, cidx);

    // 4) transposed / gathered outputs
    dim3 ogrid(DD / 32, NN / 32, BB);
    dim3 oblk(32, 8);
    write_outputs_kernel<<<ogrid, oblk, 0, stream>>>(seq1, seq2, cidx, out);
}